// WaveletRefinement_5351529250937
// MI455X (gfx1250) — compile-verified
//
#include <hip/hip_runtime.h>

typedef float v2f __attribute__((ext_vector_type(2)));
typedef float v4f __attribute__((ext_vector_type(4)));
typedef float v8f __attribute__((ext_vector_type(8)));

// GEMM: out[m][n] = act( sum_k A[m][k] * W[n][k] + bias[n] )
//   A: 8x1024 (logically padded to 16 rows with zeros), W: 1024x1024 row-major.
// One wave per 16-column tile; K-loop of V_WMMA_F32_16X16X4_F32.
// Padding handled branchlessly: clamp row address + multiply fragment by 0/1 mask.
// ACT==0: silu, ACT==1: sigmoid
template <int ACT>
__global__ __launch_bounds__(32)
void gemm8x1024_wmma(const float* __restrict__ A,
                     const float* __restrict__ W,
                     const float* __restrict__ bias,
                     float* __restrict__ out)
{
    const int lane = threadIdx.x;
    const int half = lane >> 4;     // 0: lanes 0-15, 1: lanes 16-31
    const int lidx = lane & 15;
    const int n    = blockIdx.x * 16 + lidx;   // output column for this lane

    const float* wrow  = W + (size_t)n * 1024;          // B[k][n] = W[n][k]
    const float* arow  = A + (size_t)(lidx & 7) * 1024; // always-valid address
    const float  fmask = (lidx < 8) ? 1.0f : 0.0f;      // zero for padding rows

    v8f acc = {0.f, 0.f, 0.f, 0.f, 0.f, 0.f, 0.f, 0.f};

#pragma unroll 8
    for (int k0 = 0; k0 < 1024; k0 += 4) {
        const int kA = k0 + 2 * half;   // even -> 8B-aligned float2 loads
        // A-frag (16x4 f32): VGPR0 = K=2*half, VGPR1 = K=2*half+1, row = lane%16
        v2f a = *(const v2f*)(arow + kA);
        a *= fmask;
        // B-frag (4x16 f32): same K striping, N = lane%16 within the tile
        const v2f b = *(const v2f*)(wrow + kA);
        // 8 args: (neg_a, A, neg_b, B, c_mod, C, reuse_a, reuse_b)
        acc = __builtin_amdgcn_wmma_f32_16x16x4_f32(
            false, a, false, b, (short)0, acc, false, false);
    }

    // C/D layout: VGPR r -> row r (lanes 0-15) / row r+8 (lanes 16-31), col = lane%16.
    // Rows 8..15 are padding, so only lanes 0-15 write.
    if (half == 0) {
        const float bb = bias[n];
#pragma unroll
        for (int r = 0; r < 8; ++r) {
            const float x = acc[r] + bb;
            const float s = 1.0f / (1.0f + __expf(-x));  // sigmoid(x)
            out[(size_t)r * 1024 + n] = (ACT == 0) ? (x * s) : s;
        }
    }
}

// out[b,n,d] = v[b,n,d] * (1 + gate[b,d])   -- pure HBM-bandwidth kernel.
// Non-temporal b128 load/store for the streamed v/out; gate (32 KB) stays cached.
__global__ __launch_bounds__(256)
void scale_add_kernel(const v4f* __restrict__ v4,
                      const float* __restrict__ gate,
                      v4f* __restrict__ out4,
                      int total4)
{
    const int i = blockIdx.x * 256 + threadIdx.x;
    if (i >= total4) return;

    const v4f vv = __builtin_nontemporal_load(&v4[i]);
    const unsigned e0 = (unsigned)i << 2;      // flat element index
    const unsigned b  = e0 >> 22;              // / (4096*1024)
    const unsigned dd = e0 & 1023u;            // % 1024 (4 consecutive d's, same b)
    const v4f g = *(const v4f*)(gate + ((size_t)b << 10) + dd);

    v4f o = vv * (1.0f + g);
    __builtin_nontemporal_store(o, &out4[i]);
}

extern "C" void kernel_launch(void* const* d_in, const int* in_sizes, int n_in,
                              void* d_out, int out_size, void* d_ws, size_t ws_size,
                              hipStream_t stream)
{
    const float* v      = (const float*)d_in[0];  // (8, 4096, 1024)
    const float* q_pool = (const float*)d_in[1];  // (8, 1024)
    const float* W1     = (const float*)d_in[2];  // (1024, 1024)
    const float* b1     = (const float*)d_in[3];  // (1024,)
    const float* W2     = (const float*)d_in[4];  // (1024, 1024)
    const float* b2     = (const float*)d_in[5];  // (1024,)
    float* out = (float*)d_out;

    float* h    = (float*)d_ws;            // 8*1024 floats (32 KB)
    float* gate = (float*)d_ws + 8 * 1024; // 8*1024 floats (32 KB)

    // gate = sigmoid( silu(q_pool @ W1^T + b1) @ W2^T + b2 )
    gemm8x1024_wmma<0><<<64, 32, 0, stream>>>(q_pool, W1, b1, h);
    gemm8x1024_wmma<1><<<64, 32, 0, stream>>>(h,      W2, b2, gate);

    // out = v * (1 + gate)   [wavelet pyramid collapses algebraically]
    const int total4 = 8 * 4096 * 1024 / 4;  // 8388608 float4's
    scale_add_kernel<<<total4 / 256, 256, 0, stream>>>(
        (const v4f*)v, gate, (v4f*)out, total4);
}